// Cache_83897891160781
// MI455X (gfx1250) — compile-verified
//
#include <hip/hip_runtime.h>
#include <math.h>

// Problem constants (from reference)
#define T_DIM 16
#define B_DIM 16
#define N_BLK 256
#define H_DIM 512
#define L_DIM 64
#define KC    32               // K-chunk (floats) staged to LDS per pipeline stage
#define NCHUNK (H_DIM / KC)    // 16
#define KPAD  (KC + 4)         // padded LDS row stride (words); 36 mod 64 -> conflict-free b64 reads
#define NBUF  3                // triple buffer: prefetch depth 2, per-wave pipeline
#define SSN   (KC / 4)         // 8 WMMA k-steps per chunk

typedef float v2f __attribute__((ext_vector_type(2)));
typedef float v8f __attribute__((ext_vector_type(8)));

// D = A(16x4 f32) * B(4x16 f32) + C(16x16 f32), wave32
__device__ __forceinline__ v8f wmma_f32_16x16x4(v2f a, v2f b, v8f c) {
  return __builtin_amdgcn_wmma_f32_16x16x4_f32(false, a, false, b, (short)0, c,
                                               false, false);
}

// CDNA5 async copy: 16 bytes per lane, global -> LDS, tracked by ASYNCcnt.
__device__ __forceinline__ void async_b128_to_lds(const float* gaddr, unsigned lds_off) {
  asm volatile("global_load_async_to_lds_b128 %0, %1, off"
               :: "v"(lds_off), "v"((unsigned long long)(size_t)gaddr)
               : "memory");
}

// ---------------------------------------------------------------------------
// Phase 1: scores[t][b][n] = max_j ( q[t,b,:] . keys[n,b,j,:] )
// One 4-wave workgroup per (b, n). Each wave owns a 16-row j-tile and runs a
// private triple-buffered async global->LDS pipeline (its LDS rows are
// disjoint from other waves'), so no per-chunk workgroup barriers are needed.
// Per chunk: wait async -> stage fragments to regs -> drain DS (doubles as the
// WAR fence vs the async engine) -> issue chunk c+2 -> 8 stall-free WMMAs.
// Raw scores land in the attention region of d_out; phase 2 softmaxes in place.
// ---------------------------------------------------------------------------
__global__ __launch_bounds__(128) void cache_scores_kernel(
    const float* __restrict__ q, const float* __restrict__ keys,
    float* __restrict__ scores) {
  const int wg = blockIdx.x;            // wg == n*B + b -> streams keys in memory order
  const int b  = wg & (B_DIM - 1);
  const int n  = wg >> 4;

  extern __shared__ float smem[];
  float* q2   = smem;                    // 8192 f: q transposed, pair-major q2[k/2][t][2]
  float* kbuf = smem + T_DIM * H_DIM;    // 3 buffers of 64*KPAD floats

  const int tid  = threadIdx.x;
  const int wave = tid >> 5;
  const int lane = tid & 31;
  const int tl   = lane & 15;            // M / N position within half-wave
  const int hh   = lane >> 4;            // K half-select (A/B 32-bit WMMA layout)

  const float* kbase = keys + (size_t)wg * (L_DIM * H_DIM);

  // Issue one K-chunk for this wave's 16 rows (4 async b128 per wave -> 2 KB).
  auto issue = [&](int c) {
    float* buf = kbuf + (c % NBUF) * (L_DIM * KPAD);
    const float* g = kbase + c * KC + (lane & 7) * 4;
    const unsigned lb = (unsigned)(size_t)buf + (unsigned)((lane & 7) * 16);
#pragma unroll
    for (int it = 0; it < 4; ++it) {
      const int j = wave * 16 + it * 4 + (lane >> 3);
      async_b128_to_lds(g + (size_t)j * H_DIM, lb + (unsigned)(j * KPAD * 4));
    }
  };

  issue(0);
  issue(1);  // 2 chunks in flight before q staging

  // Stage q(b) into LDS transposed: element (t,k) -> word (k/2)*32 + 2t + (k&1).
  // Hot-loop A-fragment reads are then fully bank-conflict-free.
  for (int idx = tid; idx < T_DIM * H_DIM; idx += 128) {
    const int t = idx >> 9;
    const int k = idx & (H_DIM - 1);
    const float v = q[(size_t)(t * B_DIM + b) * H_DIM + k];
    q2[(k >> 1) * 32 + t * 2 + (k & 1)] = v;
  }
  __syncthreads();                       // q2 visible to all waves (only shared LDS)

  v8f acc[4] = {};                       // 4 independent WMMA chains
  const int j0 = wave * 16;              // this wave's j-tile

#pragma unroll                           // full unroll: static buffers + wait imms
  for (int c = 0; c < NCHUNK; ++c) {
    // Chunk c resident in LDS? in-flight set is {c, c+1} except at the tail.
    if (c == NCHUNK - 1) asm volatile("s_wait_asynccnt 0" ::: "memory");
    else                 asm volatile("s_wait_asynccnt 4" ::: "memory");

    // A frag: lanes 0-15 hold (K=k,k+1) of row M=tl, lanes 16-31 hold (K=k+2,k+3).
    // B frag mirrored: lanes 0-15 = rows K=k,k+1 at N=tl; 16-31 = K=k+2,k+3.
    const float* buf  = kbuf + (c % NBUF) * (L_DIM * KPAD);
    const float* brow = buf + (j0 + tl) * KPAD + 2 * hh;
    const float* aq   = q2 + (c * (KC / 2) + hh) * 32 + tl * 2;

    v2f af[SSN], bf[SSN];
#pragma unroll
    for (int ss = 0; ss < SSN; ++ss) {
      af[ss] = *(const v2f*)(aq + ss * 64);
      bf[ss] = *(const v2f*)(brow + ss * 4);
    }
    // Fragments -> registers; also the WAR fence: every read of buffer
    // (c+2)%3 (== (c-1)%3) is now retired, so the async engine may overwrite it.
    asm volatile("s_wait_dscnt 0" ::: "memory");
    if (c + 2 < NCHUNK) issue(c + 2);

#pragma unroll
    for (int ss = 0; ss < SSN; ++ss)
      acc[ss & 3] = wmma_f32_16x16x4(af[ss], bf[ss], acc[ss & 3]);
  }
  __syncthreads();                       // all waves done reading q2 before reuse

  const v8f tot = acc[0] + acc[1] + acc[2] + acc[3];

  // Max over j: C layout VGPR r -> (M=r lanes 0-15 | M=r+8 lanes 16-31), N=tl.
  float* red = smem;                     // reuse q2 region
#pragma unroll
  for (int r = 0; r < 8; ++r) {
    float v = tot[r];
    v = fmaxf(v, __shfl_xor(v, 1));
    v = fmaxf(v, __shfl_xor(v, 2));
    v = fmaxf(v, __shfl_xor(v, 4));
    v = fmaxf(v, __shfl_xor(v, 8));
    if (tl == 0) red[wave * 16 + (r + 8 * hh)] = v;
  }
  __syncthreads();
  if (tid < 16) {  // tid == t
    const float mx = fmaxf(fmaxf(red[tid], red[16 + tid]),
                           fmaxf(red[32 + tid], red[48 + tid]));
    scores[(size_t)tid * (B_DIM * N_BLK) + b * N_BLK + n] = mx;
  }
}

// ---------------------------------------------------------------------------
// Phase 2: in-place softmax over N plus top-8 indices (jax tie-break: lower
// index first). One wave32 per (t,b) row.
// ---------------------------------------------------------------------------
__global__ __launch_bounds__(32) void softmax_topk_kernel(
    float* __restrict__ att, float* __restrict__ topk) {
  const int row  = blockIdx.x;           // t*B + b
  const int lane = threadIdx.x;
  float* arow = att + (size_t)row * N_BLK;
  const float scale = 0.044194173824159216f;  // THETA / sqrt(H)

  float s[8];
#pragma unroll
  for (int i = 0; i < 8; ++i) s[i] = arow[lane + 32 * i] * scale;

  float m = s[0];
#pragma unroll
  for (int i = 1; i < 8; ++i) m = fmaxf(m, s[i]);
#pragma unroll
  for (int d = 1; d < 32; d <<= 1) m = fmaxf(m, __shfl_xor(m, d));

  float e[8];
  float sum = 0.f;
#pragma unroll
  for (int i = 0; i < 8; ++i) { e[i] = __expf(s[i] - m); sum += e[i]; }
#pragma unroll
  for (int d = 1; d < 32; d <<= 1) sum += __shfl_xor(sum, d);
  const float inv = 1.0f / sum;
#pragma unroll
  for (int i = 0; i < 8; ++i) arow[lane + 32 * i] = e[i] * inv;

#pragma unroll 1
  for (int k = 0; k < 8; ++k) {
    float bv = -INFINITY;
    int   bi = 0x7fffffff;
#pragma unroll
    for (int i = 0; i < 8; ++i) {
      const int ix = lane + 32 * i;
      if (s[i] > bv || (s[i] == bv && ix < bi)) { bv = s[i]; bi = ix; }
    }
#pragma unroll
    for (int d = 1; d < 32; d <<= 1) {
      const float ov = __shfl_xor(bv, d);
      const int   oi = __shfl_xor(bi, d);
      if (ov > bv || (ov == bv && oi < bi)) { bv = ov; bi = oi; }
    }
    if (lane == 0) topk[k * (T_DIM * B_DIM) + row] = (float)bi;
    if ((bi & 31) == lane) s[bi >> 5] = -INFINITY;  // owner retires winner
  }
}

extern "C" void kernel_launch(void* const* d_in, const int* in_sizes, int n_in,
                              void* d_out, int out_size, void* d_ws, size_t ws_size,
                              hipStream_t stream) {
  (void)in_sizes; (void)n_in; (void)d_ws; (void)ws_size; (void)out_size;
  const float* q    = (const float*)d_in[0];   // (T,1,B,H) fp32
  const float* keys = (const float*)d_in[1];   // (N,B,L*H) fp32
  float* att  = (float*)d_out;                 // (T,B,N) attention
  float* topk = att + T_DIM * B_DIM * N_BLK;   // (k,T,B) indices (as float)

  const size_t smem_bytes =
      (size_t)(T_DIM * H_DIM + NBUF * L_DIM * KPAD) * sizeof(float);  // 60,416 B

  cache_scores_kernel<<<dim3(B_DIM * N_BLK), dim3(128), smem_bytes, stream>>>(q, keys, att);
  softmax_topk_kernel<<<dim3(T_DIM * B_DIM), dim3(32), 0, stream>>>(att, topk);
}